// PromptGuidedFusion_21775484191118
// MI455X (gfx1250) — compile-verified
//
#include <hip/hip_runtime.h>
#include <hip/hip_bf16.h>
#include <math.h>

typedef __attribute__((ext_vector_type(2))) float v2f;
typedef __attribute__((ext_vector_type(4))) float f4;
typedef __attribute__((ext_vector_type(8))) float v8f;

#define D_DIM 256
#define B_SZ  32
#define HW_SZ 4096

// ---------------------------------------------------------------------------
// Small fp32 GEMM via V_WMMA_F32_16X16X4_F32:
//   out[m][n] = act( sum_k A[m][k] * W[n][k] + bias[n] ),  M=32, N=K=256
// One wave per 16x16 output tile (grid = 2 * 16 = 32 waves).
// Per-lane WMMA layouts (CDNA5 ISA 7.12.2):
//   A (16x4, M x K):  component r holds A[m = lane&15][k0 + r + 2*(lane>=16)]
//   B (4x16, K x N):  component r holds W[n = lane&15][k0 + r + 2*(lane>=16)]
//   C/D (16x16):      vgpr r holds out[m = r + 8*(lane>=16)][n = lane&15]
// ---------------------------------------------------------------------------
__global__ __launch_bounds__(32) void pgf_gemm_wmma(
    const float* __restrict__ A,     // [32 x 256] row-major
    const float* __restrict__ W,     // [256 x 256] row-major (out-dim major)
    const float* __restrict__ bias,  // [256]
    float* __restrict__ out,         // [32 x 256] row-major
    int act_sigmoid)
{
    const int K = D_DIM, N = D_DIM;
    const int tile = blockIdx.x;        // 0..31
    const int m0 = (tile & 1) * 16;     // 2 M-tiles
    const int n0 = (tile >> 1) * 16;    // 16 N-tiles
    const int lane = threadIdx.x & 31;
    const int lo = lane & 15;
    const int hi = lane >> 4;           // 0 or 1

    const float* arow = A + (m0 + lo) * K + 2 * hi;
    const float* wrow = W + (n0 + lo) * K + 2 * hi;

    v8f acc = {};
#pragma unroll 8
    for (int k0 = 0; k0 < K; k0 += 4) {
        v2f a = *reinterpret_cast<const v2f*>(arow + k0);
        v2f b = *reinterpret_cast<const v2f*>(wrow + k0);
        // 8 args: (neg_a, A, neg_b, B, c_mod, C, reuse_a, reuse_b)
        acc = __builtin_amdgcn_wmma_f32_16x16x4_f32(
            false, a, false, b, (short)0, acc, false, false);
    }

    const int n = n0 + lo;
    const float bs = bias[n];
#pragma unroll
    for (int r = 0; r < 8; ++r) {
        const int m = m0 + r + 8 * hi;
        float v = acc[r] + bs;
        if (act_sigmoid) v = 1.0f / (1.0f + __expf(-v));
        out[m * N + n] = v;
    }
}

// ---------------------------------------------------------------------------
// Bandwidth-bound fusion: per (b, pixel) LayerNorm over 256 channels plus
// broadcast attn add and gate scaling. Each thread owns 4 consecutive pixels
// (float4 = global b128, fully coalesced). Two channel passes; pass-2 loads
// and all stores are non-temporal so the output stream does not evict the
// visual tile from L2 between the passes.
// ---------------------------------------------------------------------------
__global__ __launch_bounds__(256) void pgf_fuse(
    const float* __restrict__ visual,   // [32, 256, 64, 64]
    const float* __restrict__ attn,     // [32, 256]
    const float* __restrict__ gate,     // [32, 256]
    const float* __restrict__ ln_g,     // [256]
    const float* __restrict__ ln_b,     // [256]
    float* __restrict__ out)            // [32, 256, 64, 64]
{
    const int C = D_DIM;
    const int P4 = HW_SZ / 4;           // 1024 float4 pixels per image plane
    const int t = blockIdx.x * 256 + threadIdx.x;
    const int b = t / P4;               // uniform per block (P4 % 256 == 0)
    const int p4 = t & (P4 - 1);

    const f4* vbase = reinterpret_cast<const f4*>(visual) + (size_t)b * C * P4 + p4;
    f4*       obase = reinterpret_cast<f4*>(out)          + (size_t)b * C * P4 + p4;
    const float* arow = attn + b * C;   // wave-uniform -> scalar loads
    const float* grow = gate + b * C;

    f4 sum   = {0.f, 0.f, 0.f, 0.f};
    f4 sumsq = {0.f, 0.f, 0.f, 0.f};
#pragma unroll 4
    for (int c = 0; c < C; ++c) {
        f4 x = vbase[(size_t)c * P4];
        x = x + arow[c];
        sum   = sum + x;
        sumsq = sumsq + x * x;
    }

    const float invC = 1.0f / (float)C;
    const f4 mu  = sum * invC;
    const f4 var = sumsq * invC - mu * mu;
    f4 inv;
    inv.x = rsqrtf(var.x + 1e-5f);
    inv.y = rsqrtf(var.y + 1e-5f);
    inv.z = rsqrtf(var.z + 1e-5f);
    inv.w = rsqrtf(var.w + 1e-5f);

#pragma unroll 4
    for (int c = 0; c < C; ++c) {
        f4 x = __builtin_nontemporal_load(vbase + (size_t)c * P4);
        const float gm = 1.0f + grow[c];
        const float sc = ln_g[c] * gm;          // fold gate into LN scale
        const float sh = ln_b[c] * gm;          // ...and LN shift
        f4 y = ((x + arow[c]) - mu) * inv;
        y = y * sc + sh;
        __builtin_nontemporal_store(y, obase + (size_t)c * P4);
    }
}

extern "C" void kernel_launch(void* const* d_in, const int* in_sizes, int n_in,
                              void* d_out, int out_size, void* d_ws, size_t ws_size,
                              hipStream_t stream)
{
    const float* visual  = (const float*)d_in[0];
    const float* text    = (const float*)d_in[1];
    const float* metrics = (const float*)d_in[2];
    const float* Wv      = (const float*)d_in[3];
    const float* bv      = (const float*)d_in[4];
    const float* Wo      = (const float*)d_in[5];
    const float* bo      = (const float*)d_in[6];
    const float* ln_g    = (const float*)d_in[7];
    const float* ln_b    = (const float*)d_in[8];
    const float* gW      = (const float*)d_in[9];
    const float* gb      = (const float*)d_in[10];
    float* out = (float*)d_out;

    float* ws     = (float*)d_ws;
    float* v_proj = ws;                    // 32*256 floats
    float* attn   = ws + B_SZ * D_DIM;     // 32*256 floats
    float* gate   = ws + 2 * B_SZ * D_DIM; // 32*256 floats

    // v = text @ Wv^T + bv
    pgf_gemm_wmma<<<32, 32, 0, stream>>>(text, Wv, bv, v_proj, 0);
    // attn = v @ Wo^T + bo   (stream-ordered after the previous launch)
    pgf_gemm_wmma<<<32, 32, 0, stream>>>(v_proj, Wo, bo, attn, 0);
    // gate = sigmoid(metrics @ gW^T + gb)
    pgf_gemm_wmma<<<32, 32, 0, stream>>>(metrics, gW, gb, gate, 1);

    const int total4 = B_SZ * HW_SZ / 4;   // 32768 threads, 4 pixels each
    pgf_fuse<<<total4 / 256, 256, 0, stream>>>(visual, attn, gate, ln_g, ln_b, out);
}